// Model_9079560864162
// MI455X (gfx1250) — compile-verified
//
#include <hip/hip_runtime.h>
#include <math.h>

// ---------------- model dimensions ----------------
#define V_DIM   45004
#define VEXT_D  45054
#define EMB_D   300
#define ANS_D   16
#define H_D     512
#define HH_D    1024
#define B_D     16
#define S_D     400
#define TQ_D    50
#define TD_D    49          // TQ - 1
#define NEG_F   1e12f

typedef float v2f __attribute__((ext_vector_type(2)));
typedef float v8f __attribute__((ext_vector_type(8)));

__device__ __forceinline__ float sigf(float x) { return 1.0f / (1.0f + expf(-x)); }

__device__ __forceinline__ float wsum(float v) {
#pragma unroll
  for (int o = 16; o > 0; o >>= 1) v += __shfl_xor(v, o, 32);
  return v;
}
__device__ __forceinline__ float wmaxr(float v) {
#pragma unroll
  for (int o = 16; o > 0; o >>= 1) v = fmaxf(v, __shfl_xor(v, o, 32));
  return v;
}

// =====================================================================
// Generic fp32 WMMA GEMM:  C = act( A1 @ B1^T(+) + A2 @ B2^T(+) + bias )
//   TRANSB=1 : B indexed [n][k]  (weight-style, matches "x @ W.T")
//   TRANSB=0 : B indexed [k][n]
// Each wave computes a 16(M) x 64(N) strip; block = 8 waves = 16x512 of C.
// A/B fragments per ISA 7.12.2: lane<16 -> K{k,k+1}, lane>=16 -> K{k+2,k+3}.
// =====================================================================
template <int TRANSB>
__global__ void __launch_bounds__(256)
gemm_wmma(const float* __restrict__ A1, long lda1, const float* __restrict__ B1, long ldb1, int K1,
          const float* __restrict__ A2, long lda2, const float* __restrict__ B2, long ldb2, int K2,
          const float* __restrict__ bias, float* __restrict__ C, long ldc,
          int M, int N, long sA1, long sA2, long sB1, long sB2, long sC, int act)
{
  const int wv   = threadIdx.x >> 5;
  const int lane = threadIdx.x & 31;
  const int hf   = lane >> 4;       // half selector (K += 2)
  const int lm   = lane & 15;
  const int m0   = blockIdx.y << 4;
  const int n0   = (blockIdx.x << 9) + (wv << 6);
  const long z   = blockIdx.z;
  A1 += z * sA1; B1 += z * sB1; C += z * sC;
  if (A2) { A2 += z * sA2; B2 += z * sB2; }

  int mrow = m0 + lm; if (mrow > M - 1) mrow = M - 1;
  v8f acc[4] = {};

#pragma unroll 1
  for (int pass = 0; pass < 2; ++pass) {
    const float* A = pass ? A2 : A1;
    if (!A) continue;
    const float* B = pass ? B2 : B1;
    const long lda = pass ? lda2 : lda1;
    const long ldb = pass ? ldb2 : ldb1;
    const int  K   = pass ? K2  : K1;   // must be a multiple of 4 (true for all uses)
    const float* ar = A + (long)mrow * lda + 2 * hf;
    for (int k = 0; k < K; k += 4) {
      v2f a; a.x = ar[k]; a.y = ar[k + 1];
#pragma unroll
      for (int j = 0; j < 4; ++j) {
        int nt = n0 + (j << 4);
        if (nt < N) {                       // wave-uniform -> EXEC stays all-ones
          int n = nt + lm; if (n > N - 1) n = N - 1;   // clamp: garbage cols never stored
          v2f b;
          if (TRANSB) {
            const float* br = B + (long)n * ldb + k + 2 * hf;
            b.x = br[0]; b.y = br[1];
          } else {
            b.x = B[(long)(k + 2 * hf)     * ldb + n];
            b.y = B[(long)(k + 2 * hf + 1) * ldb + n];
          }
          acc[j] = __builtin_amdgcn_wmma_f32_16x16x4_f32(false, a, false, b, (short)0,
                                                         acc[j], false, false);
        }
      }
    }
  }

#pragma unroll
  for (int j = 0; j < 4; ++j) {
    int nt = n0 + (j << 4);
    if (nt < N) {
      int n = nt + lm;
      if (n < N) {
        float bv = bias ? bias[n] : 0.0f;
#pragma unroll
        for (int vv = 0; vv < 8; ++vv) {
          int m = m0 + vv + 8 * hf;        // D layout: VGPR v -> row v + 8*half
          if (m < M) {
            float val = acc[j][vv] + bv;
            if (act == 1)      val = tanhf(val);
            else if (act == 2) val = sigf(val);
            C[(long)m * ldc + n] = val;
          }
        }
      }
    }
  }
}

// =====================================================================
// Encoder LSTM recurrence, both directions in one launch (block 0 = fwd,
// block 1 = bwd), 1024 threads = 32 waves = one WGP per direction.
// Wave w owns hidden slice [16w,16w+16): its i/f/g/o gate tiles live in 4
// WMMA accumulators, c stays in VGPRs, h is shared via padded LDS.
// gX holds the precomputed x@Wih^T + b, laid out (b, t, 2048).
// =====================================================================
__global__ void __launch_bounds__(1024)
lstm_pair(const float* __restrict__ gXf, const float* __restrict__ gXb,
          const float* __restrict__ Whhf, const float* __restrict__ Whhb,
          const int* __restrict__ pid,
          float* __restrict__ outBuf, float* __restrict__ hT, float* __restrict__ cT)
{
  const int dir = blockIdx.x;
  const float* gX  = dir ? gXb  : gXf;
  const float* Whh = dir ? Whhb : Whhf;
  const int off = dir ? H_D : 0;

  __shared__ float hS[16 * 516];           // 16 x 512, stride 516 -> conflict-free
  const int tid = threadIdx.x;
  const int wv = tid >> 5, lane = tid & 31, hf = lane >> 4, lm = lane & 15;
  const int jb = wv << 4;
  const int j  = jb + lm;

  for (int i = tid; i < 16 * 516; i += 1024) hS[i] = 0.0f;
  float cR[8];
#pragma unroll
  for (int vv = 0; vv < 8; ++vv) cR[vv] = 0.0f;
  __syncthreads();

  const float* Wi = Whh + (long)(jb        + lm) * H_D + 2 * hf;
  const float* Wf = Whh + (long)(jb +  512 + lm) * H_D + 2 * hf;
  const float* Wg = Whh + (long)(jb + 1024 + lm) * H_D + 2 * hf;
  const float* Wo = Whh + (long)(jb + 1536 + lm) * H_D + 2 * hf;
  const float* hr = &hS[lm * 516 + 2 * hf];

  for (int tt = 0; tt < S_D; ++tt) {
    const int t = dir ? (S_D - 1 - tt) : tt;
    v8f ai = {}, af = {}, ag = {}, ao = {};
    for (int k = 0; k < H_D; k += 4) {
      v2f a; a.x = hr[k]; a.y = hr[k + 1];
      v2f b;
      b.x = Wi[k]; b.y = Wi[k + 1];
      ai = __builtin_amdgcn_wmma_f32_16x16x4_f32(false, a, false, b, (short)0, ai, false, false);
      b.x = Wf[k]; b.y = Wf[k + 1];
      af = __builtin_amdgcn_wmma_f32_16x16x4_f32(false, a, false, b, (short)0, af, false, false);
      b.x = Wg[k]; b.y = Wg[k + 1];
      ag = __builtin_amdgcn_wmma_f32_16x16x4_f32(false, a, false, b, (short)0, ag, false, false);
      b.x = Wo[k]; b.y = Wo[k + 1];
      ao = __builtin_amdgcn_wmma_f32_16x16x4_f32(false, a, false, b, (short)0, ao, false, false);
    }
    __syncthreads();                        // all hS reads done before updates
#pragma unroll
    for (int vv = 0; vv < 8; ++vv) {
      const int m = vv + 8 * hf;            // batch row (D-matrix layout)
      const long rx = ((long)m * S_D + t) * 2048;
      float iv = ai[vv] + gX[rx +        j];
      float fv = af[vv] + gX[rx +  512 + j];
      float gv = ag[vv] + gX[rx + 1024 + j];
      float ov = ao[vv] + gX[rx + 1536 + j];
      float c  = cR[vv];
      float cn = sigf(fv) * c + sigf(iv) * tanhf(gv);
      float hn = sigf(ov) * tanhf(cn);
      const bool mk = pid[m * S_D + t] != 0;
      float hold = hS[m * 516 + j];
      cR[vv]          = mk ? cn : c;
      hS[m * 516 + j] = mk ? hn : hold;
      outBuf[((long)m * S_D + t) * HH_D + off + j] = mk ? hn : 0.0f;
    }
    __syncthreads();
  }
#pragma unroll
  for (int vv = 0; vv < 8; ++vv) {
    const int m = vv + 8 * hf;
    hT[m * HH_D + off + j] = hS[m * 516 + j];
    cT[m * HH_D + off + j] = cR[vv];
  }
}

// ---------------- embedding + layernorm (wave per token) ----------------
__device__ __forceinline__ void ln_row(const float* __restrict__ row, const float* __restrict__ g,
                                       const float* __restrict__ b, float* __restrict__ out, int lane)
{
  float x[10]; float s = 0.f;
#pragma unroll
  for (int q = 0; q < 10; ++q) { int k = lane + 32 * q; x[q] = (k < EMB_D) ? row[k] : 0.f; s += x[q]; }
  s = wsum(s);
  float mu = s * (1.0f / EMB_D);
  float vs = 0.f;
#pragma unroll
  for (int q = 0; q < 10; ++q) { int k = lane + 32 * q; if (k < EMB_D) { float d = x[q] - mu; vs += d * d; } }
  vs = wsum(vs);
  float inv = 1.0f / sqrtf(vs * (1.0f / EMB_D) + 1e-12f);
#pragma unroll
  for (int q = 0; q < 10; ++q) { int k = lane + 32 * q; if (k < EMB_D) out[k] = g[k] * (x[q] - mu) * inv + b[k]; }
}

__global__ void __launch_bounds__(256)
embed_para(const int* __restrict__ pid, const int* __restrict__ ans,
           const float* __restrict__ wemb, const float* __restrict__ aemb,
           const float* __restrict__ lg, const float* __restrict__ lb, float* __restrict__ e)
{
  const int wv = threadIdx.x >> 5, lane = threadIdx.x & 31;
  const int r = blockIdx.x * 8 + wv;
  if (r >= B_D * S_D) return;
  ln_row(wemb + (long)pid[r] * EMB_D, lg, lb, e + (long)r * 316, lane);
  if (lane < ANS_D) e[(long)r * 316 + EMB_D + lane] = aemb[ans[r] * ANS_D + lane];
}

__global__ void __launch_bounds__(256)
embed_q(const int* __restrict__ qid, const float* __restrict__ wemb,
        const float* __restrict__ lg, const float* __restrict__ lb, float* __restrict__ demb)
{
  const int wv = threadIdx.x >> 5, lane = threadIdx.x & 31;
  const int r = blockIdx.x * 8 + wv;
  if (r >= B_D * TD_D) return;
  const int b = r / TD_D, t = r % TD_D;
  ln_row(wemb + (long)qid[b * TQ_D + t] * EMB_D, lg, lb, demb + (long)r * EMB_D, lane);
}

// ---------------- masked row softmax (wave per row of 400) ----------------
__global__ void __launch_bounds__(256)
softmax_rows(float* __restrict__ en, const int* __restrict__ pid)
{
  const int wv = threadIdx.x >> 5, lane = threadIdx.x & 31;
  const int r = blockIdx.x * 8 + wv;             // r = b*400 + q, 6400 rows exact
  float* row = en + (long)r * S_D;
  const int* pm = pid + (r / S_D) * S_D;
  float mx = -3.0e38f;
  for (int k = lane; k < S_D; k += 32) {
    float v = (pm[k] != 0) ? row[k] : -NEG_F;
    row[k] = v;
    mx = fmaxf(mx, v);
  }
  mx = wmaxr(mx);
  float s = 0.f;
  for (int k = lane; k < S_D; k += 32) s += expf(row[k] - mx);
  s = wsum(s);
  float inv = 1.0f / s;
  for (int k = lane; k < S_D; k += 32) row[k] = expf(row[k] - mx) * inv;
}

// ---------------- misc elementwise ----------------
__global__ void __launch_bounds__(256)
enc_combine(const float* __restrict__ f, const float* __restrict__ g,
            const float* __restrict__ o, float* __restrict__ out, long n)
{
  long i = (long)blockIdx.x * 256 + threadIdx.x;
  if (i < n) { float gv = g[i]; out[i] = gv * f[i] + (1.0f - gv) * o[i]; }
}

__global__ void __launch_bounds__(256)
fill_zero(float* __restrict__ p, long n)
{
  long i = (long)blockIdx.x * 256 + threadIdx.x;
  if (i < n) p[i] = 0.0f;
}

// decoder LSTM cell pointwise: gates (16 x 4096) -> update h,c (16 x 1024)
__global__ void __launch_bounds__(256)
cell_update(const float* __restrict__ g, float* __restrict__ h, float* __restrict__ c)
{
  const int i = blockIdx.x * 256 + threadIdx.x;   // < 16*1024 (grid sized exactly)
  const int m = i >> 10, jj = i & 1023;
  const float* gr = g + (long)m * 4096;
  float iv = gr[jj], fv = gr[HH_D + jj], gv = gr[2 * HH_D + jj], ov = gr[3 * HH_D + jj];
  float cv = c[i];
  float cn = sigf(fv) * cv + sigf(iv) * tanhf(gv);
  c[i] = cn;
  h[i] = sigf(ov) * tanhf(cn);
}

// ---------------- decoder attention (block per batch row) ----------------
__global__ void __launch_bounds__(256)
dec_attn(const float* __restrict__ h1, const float* __restrict__ mems,
         const int* __restrict__ pid, float* __restrict__ pc,
         float* __restrict__ en2, int t)
{
  const int b = blockIdx.x;
  __shared__ float es[S_D];
  __shared__ float redA[8], redB[8];
  const int tid = threadIdx.x, wv = tid >> 5, lane = tid & 31;
  const float* hb = h1 + b * HH_D;
  const float* mb = mems + (long)b * S_D * HH_D;

  for (int s = wv; s < S_D; s += 8) {
    const float* mr = mb + (long)s * HH_D;
    float p = 0.f;
    for (int k = lane; k < HH_D; k += 32) p += hb[k] * mr[k];
    p = wsum(p);
    if (lane == 0) {
      float ev = (pid[b * S_D + s] != 0) ? p : -NEG_F;
      es[s] = ev;
      en2[((long)b * TD_D + t) * S_D + s] = ev;   // masked energy for pointer-max
    }
  }
  __syncthreads();
  float mx = -3.0e38f;
  for (int s = tid; s < S_D; s += 256) mx = fmaxf(mx, es[s]);
  mx = wmaxr(mx);
  if (lane == 0) redA[wv] = mx;
  __syncthreads();
  mx = redA[0];
#pragma unroll
  for (int q = 1; q < 8; ++q) mx = fmaxf(mx, redA[q]);
  float ls = 0.f;
  for (int s = tid; s < S_D; s += 256) ls += expf(es[s] - mx);
  ls = wsum(ls);
  if (lane == 0) redB[wv] = ls;
  __syncthreads();
  float tot = 0.f;
#pragma unroll
  for (int q = 0; q < 8; ++q) tot += redB[q];
  float inv = 1.0f / tot;
  for (int s = tid; s < S_D; s += 256) es[s] = expf(es[s] - mx) * inv;
  __syncthreads();
  for (int d = tid; d < HH_D; d += 256) {
    float acc = 0.f;
    for (int s = 0; s < S_D; ++s) acc += es[s] * mb[(long)s * HH_D + d];
    pc[b * HH_D + d] = acc;   // ctx2 -> carried as prev_context
  }
}

// ---------------- pointer scatter-max + final masking (block per (b,t)) ----------------
__global__ void __launch_bounds__(256)
pointer_final(float* __restrict__ out, const float* __restrict__ en2, const int* __restrict__ ext)
{
  const int r = blockIdx.x;             // b*TD + t
  const int b = r / TD_D;
  float* orow = out + (long)r * VEXT_D; // row already holds extd = logits(+bias) in [0,V)
  const float* er = en2 + (long)r * S_D;
  const int* ex = ext + b * S_D;

  for (int v = V_DIM + threadIdx.x; v < VEXT_D; v += 256) orow[v] = 0.0f;  // extd==0 for OOV
  __syncthreads();
  for (int s = threadIdx.x; s < S_D; s += 256) {
    int v = ex[s];
    bool owner = true;
    for (int s2 = 0; s2 < s; ++s2) if (ex[s2] == v) { owner = false; break; }
    if (!owner) continue;
    float mxv = er[s];
    for (int s2 = s + 1; s2 < S_D; ++s2) if (ex[s2] == v) mxv = fmaxf(mxv, er[s2]);
    if (mxv != -NEG_F) orow[v] += mxv;  // sm==-NEG -> 0 (skip); unique v per owner
  }
  __syncthreads();
  for (int v = threadIdx.x; v < VEXT_D; v += 256) {
    float lg = orow[v];
    orow[v] = (lg == 0.0f) ? -NEG_F : lg;
  }
}

// =====================================================================
// host-side launcher
// =====================================================================
static void gemmL(hipStream_t st, int transB,
                  const float* A1, long lda1, const float* B1, long ldb1, int K1,
                  const float* A2, long lda2, const float* B2, long ldb2, int K2,
                  const float* bias, float* C, long ldc, int M, int N,
                  int batch = 1, long sA1 = 0, long sA2 = 0, long sB1 = 0, long sB2 = 0,
                  long sC = 0, int act = 0)
{
  dim3 gr((unsigned)((N + 511) / 512), (unsigned)(M / 16), (unsigned)batch);
  if (transB)
    gemm_wmma<1><<<gr, 256, 0, st>>>(A1, lda1, B1, ldb1, K1, A2, lda2, B2, ldb2, K2,
                                     bias, C, ldc, M, N, sA1, sA2, sB1, sB2, sC, act);
  else
    gemm_wmma<0><<<gr, 256, 0, st>>>(A1, lda1, B1, ldb1, K1, A2, lda2, B2, ldb2, K2,
                                     bias, C, ldc, M, N, sA1, sA2, sB1, sB2, sC, act);
}

extern "C" void kernel_launch(void* const* d_in, const int* in_sizes, int n_in,
                              void* d_out, int out_size, void* d_ws, size_t ws_size,
                              hipStream_t stream)
{
  (void)in_sizes; (void)n_in; (void)out_size; (void)ws_size;
  const int* pid = (const int*)d_in[0];
  const int* ans = (const int*)d_in[1];
  const int* qid = (const int*)d_in[2];
  const int* ext = (const int*)d_in[3];
  const float* wemb    = (const float*)d_in[4];
  const float* aemb    = (const float*)d_in[5];
  const float* ln_g    = (const float*)d_in[6];
  const float* ln_b    = (const float*)d_in[7];
  const float* l0f_Wih = (const float*)d_in[8];
  const float* l0f_Whh = (const float*)d_in[9];
  const float* l0f_b   = (const float*)d_in[10];
  const float* l0b_Wih = (const float*)d_in[11];
  const float* l0b_Whh = (const float*)d_in[12];
  const float* l0b_b   = (const float*)d_in[13];
  const float* l1f_Wih = (const float*)d_in[14];
  const float* l1f_Whh = (const float*)d_in[15];
  const float* l1f_b   = (const float*)d_in[16];
  const float* l1b_Wih = (const float*)d_in[17];
  const float* l1b_Whh = (const float*)d_in[18];
  const float* l1b_b   = (const float*)d_in[19];
  const float* trans_W = (const float*)d_in[20];
  const float* trans_b = (const float*)d_in[21];
  const float* upd_W   = (const float*)d_in[22];
  const float* gate_W  = (const float*)d_in[23];
  const float* encf_W  = (const float*)d_in[24];
  const float* encf_b  = (const float*)d_in[25];
  const float* red_W   = (const float*)d_in[26];
  const float* red_b   = (const float*)d_in[27];
  const float* d0_Wih  = (const float*)d_in[28];
  const float* d0_Whh  = (const float*)d_in[29];
  const float* d0_b    = (const float*)d_in[30];
  const float* d1_Wih  = (const float*)d_in[31];
  const float* d1_Whh  = (const float*)d_in[32];
  const float* d1_b    = (const float*)d_in[33];
  const float* cat_W   = (const float*)d_in[34];
  const float* cat_b   = (const float*)d_in[35];
  const float* log_W   = (const float*)d_in[36];
  const float* log_b   = (const float*)d_in[37];
  float* out = (float*)d_out;

  // ---- workspace partition (floats) ----
  float* w = (float*)d_ws;
  auto take = [&](long n) { float* p = w; w += n; return p; };
  const long NT = (long)B_D * S_D;                 // 6400 tokens
  float* e    = take(NT * 316);
  float* demb = take((long)B_D * TD_D * EMB_D);
  float* gxf  = take(NT * 2048);
  float* gxb  = take(NT * 2048);
  float* x1   = take(NT * HH_D);
  float* outp = take(NT * HH_D);
  float* memb = take(NT * HH_D);
  float* en   = take((long)B_D * S_D * S_D);
  float* mems = take(NT * HH_D);
  float* en2  = take((long)B_D * TD_D * S_D);
  float* li   = take((long)B_D * TD_D * HH_D);
  float* h0T  = take(B_D * HH_D);
  float* c0T  = take(B_D * HH_D);
  float* h1T  = take(B_D * HH_D);
  float* c1T  = take(B_D * HH_D);
  float* pc   = take(B_D * HH_D);
  float* xbuf = take(B_D * EMB_D);
  float* g0   = take((long)B_D * 4096);
  float* g1   = take((long)B_D * 4096);
  // aliases: encoder x-gate buffers are dead after the recurrences
  float* ctx = gxf;
  float* fb  = gxf + NT * HH_D;
  float* gb  = gxb;
  float* eo  = gxb + NT * HH_D;

  // 1) embeddings + layernorm
  embed_para<<<(unsigned)(NT / 8), 256, 0, stream>>>(pid, ans, wemb, aemb, ln_g, ln_b, e);
  embed_q<<<(B_D * TD_D) / 8, 256, 0, stream>>>(qid, wemb, ln_g, ln_b, demb);

  // 2) encoder layer 0: precomputed input gates + bidirectional recurrence
  gemmL(stream, 1, e, 316, l0f_Wih, 316, 316, nullptr, 0, nullptr, 0, 0, l0f_b, gxf, 2048, (int)NT, 2048);
  gemmL(stream, 1, e, 316, l0b_Wih, 316, 316, nullptr, 0, nullptr, 0, 0, l0b_b, gxb, 2048, (int)NT, 2048);
  lstm_pair<<<2, 1024, 0, stream>>>(gxf, gxb, l0f_Whh, l0b_Whh, pid, x1, h0T, c0T);

  // 3) encoder layer 1
  gemmL(stream, 1, x1, HH_D, l1f_Wih, HH_D, HH_D, nullptr, 0, nullptr, 0, 0, l1f_b, gxf, 2048, (int)NT, 2048);
  gemmL(stream, 1, x1, HH_D, l1b_Wih, HH_D, HH_D, nullptr, 0, nullptr, 0, 0, l1b_b, gxb, 2048, (int)NT, 2048);
  lstm_pair<<<2, 1024, 0, stream>>>(gxf, gxb, l1f_Whh, l1b_Whh, pid, outp, h1T, c1T);

  // 4) encoder self-attention + gated update
  gemmL(stream, 1, outp, HH_D, trans_W, HH_D, HH_D, nullptr, 0, nullptr, 0, 0, trans_b, memb, HH_D, (int)NT, HH_D);
  gemmL(stream, 1, outp, HH_D, memb, HH_D, HH_D, nullptr, 0, nullptr, 0, 0, nullptr, en, S_D, S_D, S_D,
        B_D, (long)S_D * HH_D, 0, (long)S_D * HH_D, 0, (long)S_D * S_D);
  softmax_rows<<<(unsigned)(NT / 8), 256, 0, stream>>>(en, pid);
  gemmL(stream, 0, en, S_D, outp, HH_D, S_D, nullptr, 0, nullptr, 0, 0, nullptr, ctx, HH_D, S_D, HH_D,
        B_D, (long)S_D * S_D, 0, (long)S_D * HH_D, 0, (long)S_D * HH_D);
  gemmL(stream, 1, outp, HH_D, upd_W, 2048, HH_D, ctx, HH_D, upd_W + HH_D, 2048, HH_D,
        nullptr, fb, HH_D, (int)NT, HH_D, 1, 0, 0, 0, 0, 0, /*tanh*/1);
  gemmL(stream, 1, outp, HH_D, gate_W, 2048, HH_D, ctx, HH_D, gate_W + HH_D, 2048, HH_D,
        nullptr, gb, HH_D, (int)NT, HH_D, 1, 0, 0, 0, 0, 0, /*sigmoid*/2);
  enc_combine<<<(unsigned)((NT * HH_D) / 256), 256, 0, stream>>>(fb, gb, outp, eo, NT * HH_D);
  gemmL(stream, 1, eo, HH_D, encf_W, HH_D, HH_D, nullptr, 0, nullptr, 0, 0, encf_b, mems, HH_D, (int)NT, HH_D);

  // 5) decoder (49 sequential steps; heavy vocab GEMM hoisted out)
  fill_zero<<<(B_D * HH_D) / 256, 256, 0, stream>>>(pc, (long)B_D * HH_D);
  for (int t = 0; t < TD_D; ++t) {
    gemmL(stream, 1, demb + (long)t * EMB_D, (long)TD_D * EMB_D, red_W, 1324, EMB_D,
          pc, HH_D, red_W + EMB_D, 1324, HH_D, red_b, xbuf, EMB_D, B_D, EMB_D);
    gemmL(stream, 1, xbuf, EMB_D, d0_Wih, EMB_D, EMB_D,
          h0T, HH_D, d0_Whh, HH_D, HH_D, d0_b, g0, 4096, B_D, 4096);
    cell_update<<<64, 256, 0, stream>>>(g0, h0T, c0T);
    gemmL(stream, 1, h0T, HH_D, d1_Wih, HH_D, HH_D,
          h1T, HH_D, d1_Whh, HH_D, HH_D, d1_b, g1, 4096, B_D, 4096);
    cell_update<<<64, 256, 0, stream>>>(g1, h1T, c1T);
    dec_attn<<<B_D, 256, 0, stream>>>(h1T, mems, pid, pc, en2, t);
    gemmL(stream, 1, h1T, HH_D, cat_W, 2048, HH_D,
          pc, HH_D, cat_W + HH_D, 2048, HH_D, cat_b, li + (long)t * HH_D, (long)TD_D * HH_D,
          B_D, HH_D, 1, 0, 0, 0, 0, 0, /*tanh*/1);
  }

  // 6) vocab projection (logit_W read once, L2-resident) + pointer max
  gemmL(stream, 1, li, HH_D, log_W, HH_D, HH_D, nullptr, 0, nullptr, 0, 0,
        log_b, out, VEXT_D, B_D * TD_D, V_DIM);
  pointer_final<<<B_D * TD_D, 256, 0, stream>>>(out, en2, ext);
}